// SemanticAggregator_83640193122816
// MI455X (gfx1250) — compile-verified
//
#include <hip/hip_runtime.h>
#include <hip/hip_bf16.h>
#include <math.h>

// SemanticAggregator for MI455X (gfx1250).
//
// Math simplification: scores[b,n] = act[b,n,:]·(W^T q) + (proj_b·q); the
// second term is constant over n and cancels inside softmax, so the
// [B*N,1024]x[1024,1024] projection GEMM collapses to a dot with the
// precomputed 1024-vector qw = W^T q. The kernel is then bandwidth bound:
// 512MB activation stream + 32MB output ~= 23us at 23.3 TB/s.
//
// Scores use the fp32 tensor core path V_WMMA_F32_16X16X4_F32 (full fp32
// precision, required for the softmax over scores with sigma~18).

constexpr int kH = 1024;   // hidden dim
constexpr int kN = 16;     // num texts
constexpr float kEps = 1e-5f;

typedef float v2f __attribute__((ext_vector_type(2)));
typedef float v8f __attribute__((ext_vector_type(8)));

// ---------------------------------------------------------------------------
// Kernel 1: qw[h] = sum_d q[d] * W[d,h].  W is [H,H] row-major (d major), so
// consecutive threads (consecutive h) read consecutive addresses: coalesced.
// 4MB read total — trivial.
// ---------------------------------------------------------------------------
__global__ void qw_kernel(const float* __restrict__ W,
                          const float* __restrict__ q,
                          float* __restrict__ qw) {
  const int h = blockIdx.x * blockDim.x + threadIdx.x;
  if (h >= kH) return;
  float acc = 0.f;
#pragma unroll 8
  for (int d = 0; d < kH; ++d) acc = fmaf(q[d], W[d * kH + h], acc);
  qw[h] = acc;
}

// ---------------------------------------------------------------------------
// Kernel 2: one batch row per block; 256 threads = 8 waves (wave32).
//
// Phase 1 (scores, WMMA): wave w owns K range [w*128, w*128+128).
//   A tile (16x4 fp32): lane j supplies A[j&15, koff + {0,1}] where
//   koff = 2*(j>=16)  -> one global_load_b64 per K-step.
//   B tile (4x16): qw[k] replicated across all 16 columns, same K mapping.
//   D columns are then all identical = per-wave partial score of row M.
//   Two accumulators break the WMMA D->C dependency chain.
// Phase 2: 8-wave partials reduced in LDS; mask applied; 16-way softmax
//   computed redundantly per thread (16 values, negligible).
// Phase 3: weighted pool. Thread t owns h = [4t,4t+4): coalesced float4
//   re-read of the same 64KB tile (WGP$/L2 hit), FMA with attn weights.
// Phase 4: LayerNorm over H=1024 via shuffle + LDS block reduction.
// ---------------------------------------------------------------------------
__global__ __launch_bounds__(256)
void pool_ln_kernel(const float* __restrict__ act,
                    const float* __restrict__ qw,
                    const unsigned char* __restrict__ mask,
                    const float* __restrict__ gamma,
                    const float* __restrict__ beta,
                    float* __restrict__ out) {
  const int b    = blockIdx.x;
  const int tid  = threadIdx.x;
  const int lane = tid & 31;
  const int wave = tid >> 5;

  __shared__ float s_scores[8 * kN];  // per-wave partial scores
  __shared__ float s_final[kN];       // reduced scores
  __shared__ float s_red[8 * 2];      // layernorm {sum, sumsq} per wave

  const float* actb = act + (size_t)b * (kN * kH);

  // ---- Phase 1: scores via V_WMMA_F32_16X16X4_F32 --------------------------
  const int row   = lane & 15;          // A-matrix M index (= text index n)
  const int khalf = (lane >> 4) << 1;   // 0 for lanes 0-15, 2 for lanes 16-31
  const float* arow = actb + row * kH + khalf;
  const float* qrow = qw + khalf;

  v8f c0 = {0.f, 0.f, 0.f, 0.f, 0.f, 0.f, 0.f, 0.f};
  v8f c1 = {0.f, 0.f, 0.f, 0.f, 0.f, 0.f, 0.f, 0.f};
  const int kbase = wave * 128;
#pragma unroll 4
  for (int i = 0; i < 32; i += 2) {
    const int ka = kbase + i * 4;
    const int kb = ka + 4;
    v2f a0 = *(const v2f*)(arow + ka);
    v2f b0 = *(const v2f*)(qrow + ka);
    v2f a1 = *(const v2f*)(arow + kb);
    v2f b1 = *(const v2f*)(qrow + kb);
    c0 = __builtin_amdgcn_wmma_f32_16x16x4_f32(false, a0, false, b0,
                                               (short)0, c0, false, false);
    c1 = __builtin_amdgcn_wmma_f32_16x16x4_f32(false, a1, false, b1,
                                               (short)0, c1, false, false);
  }
  // D layout: lane 0 holds D[r,0] (rows 0-7) in its 8 VGPRs, lane 16 holds
  // rows 8-15. All columns identical (B replicated), so column 0 suffices.
  if (lane == 0) {
#pragma unroll
    for (int r = 0; r < 8; ++r) s_scores[wave * kN + r] = c0[r] + c1[r];
  }
  if (lane == 16) {
#pragma unroll
    for (int r = 0; r < 8; ++r) s_scores[wave * kN + 8 + r] = c0[r] + c1[r];
  }
  __syncthreads();

  if (tid < kN) {
    float s = 0.f;
#pragma unroll
    for (int w = 0; w < 8; ++w) s += s_scores[w * kN + tid];
    if (!mask[(size_t)b * kN + tid]) s = -INFINITY;
    s_final[tid] = s;
  }
  __syncthreads();

  // ---- Phase 2: softmax over 16, computed redundantly per thread -----------
  float sc[kN];
  float mx = -INFINITY;
#pragma unroll
  for (int n = 0; n < kN; ++n) {
    sc[n] = s_final[n];
    mx = fmaxf(mx, sc[n]);
  }
  float denom = 0.f;
#pragma unroll
  for (int n = 0; n < kN; ++n) {
    const float e = __expf(sc[n] - mx);
    sc[n] = e;
    denom += e;
  }
  const float inv = 1.f / denom;

  // ---- Phase 3: weighted pool; thread t owns h in [4t, 4t+4) ---------------
  float wx = 0.f, wy = 0.f, wz = 0.f, ww = 0.f;
  const float4* ap = (const float4*)actb + tid;  // + n*(kH/4) per text
#pragma unroll
  for (int n = 0; n < kN; ++n) {
    const float4 a4 = ap[n * (kH / 4)];
    const float wgt = sc[n] * inv;
    wx = fmaf(wgt, a4.x, wx);
    wy = fmaf(wgt, a4.y, wy);
    wz = fmaf(wgt, a4.z, wz);
    ww = fmaf(wgt, a4.w, ww);
  }

  // ---- Phase 4: LayerNorm over H=1024 --------------------------------------
  float sum = wx + wy + wz + ww;
  float ssq = wx * wx + wy * wy + wz * wz + ww * ww;
#pragma unroll
  for (int off = 16; off > 0; off >>= 1) {
    sum += __shfl_xor(sum, off, 32);
    ssq += __shfl_xor(ssq, off, 32);
  }
  if (lane == 0) {
    s_red[wave * 2 + 0] = sum;
    s_red[wave * 2 + 1] = ssq;
  }
  __syncthreads();
  float tot = 0.f, totsq = 0.f;
#pragma unroll
  for (int w = 0; w < 8; ++w) {
    tot   += s_red[w * 2 + 0];
    totsq += s_red[w * 2 + 1];
  }
  const float mu   = tot * (1.f / (float)kH);
  const float var  = totsq * (1.f / (float)kH) - mu * mu;
  const float rstd = rsqrtf(var + kEps);

  const float4 g4 = ((const float4*)gamma)[tid];
  const float4 b4 = ((const float4*)beta)[tid];
  float4 o4;
  o4.x = (wx - mu) * rstd * g4.x + b4.x;
  o4.y = (wy - mu) * rstd * g4.y + b4.y;
  o4.z = (wz - mu) * rstd * g4.z + b4.z;
  o4.w = (ww - mu) * rstd * g4.w + b4.w;
  ((float4*)(out + (size_t)b * kH))[tid] = o4;
}

// ---------------------------------------------------------------------------
extern "C" void kernel_launch(void* const* d_in, const int* in_sizes, int n_in,
                              void* d_out, int out_size, void* d_ws, size_t ws_size,
                              hipStream_t stream) {
  const float* act = (const float*)d_in[0];   // [B, 16, 1024] fp32
  const float* W   = (const float*)d_in[1];   // [1024, 1024]  fp32
  // d_in[2] = proj_b: contributes a constant (b.q) to every score, which
  // cancels inside softmax -> unused.
  const float* q     = (const float*)d_in[3]; // [1024]
  const float* gamma = (const float*)d_in[4]; // [1024]
  const float* beta  = (const float*)d_in[5]; // [1024]
  const unsigned char* mask = (const unsigned char*)d_in[6]; // [B,16] bool
  float* out = (float*)d_out;
  float* qw  = (float*)d_ws;                  // 1024 floats of scratch

  const int batch = in_sizes[0] / (kN * kH);  // 8192

  qw_kernel<<<kH / 256, 256, 0, stream>>>(W, q, qw);
  pool_ln_kernel<<<batch, 256, 0, stream>>>(act, qw, mask, gamma, beta, out);
}